// MeanAggregator_51642686767895
// MI455X (gfx1250) — compile-verified
//
#include <hip/hip_runtime.h>
#include <stdint.h>

// MeanAggregator: out[n, :] = mean_s features[neigh_idx[n, s], :]
// N_NODES=100000, D_FEAT=128 fp32, NUM_SAMPLE=10.
//
// Memory-bound gather. The 51.2 MB feature table fits in MI455X's 192 MB L2,
// so the gather stream (512 MB) is served at L2 bandwidth; HBM only sees
// table + idx + out (~110 MB -> ~5us at 23.3 TB/s). Strategy:
//   1) load all 10 neighbor indices per node in one clause (single wait),
//   2) issue all 10 CDNA5 async global->LDS B128 gathers back-to-back
//      (ASYNCcnt tracks them; zero destination VGPRs consumed),
//   3) s_wait_asynccnt 0, reduce the staged rows from LDS, NT-store result
//      so the feature table stays L2-resident.

#define D_FEAT 128
#define NUM_SAMPLE 10
#define WAVES_PER_BLOCK 8
#define THREADS (WAVES_PER_BLOCK * 32)

typedef float v4f __attribute__((ext_vector_type(4)));

// CDNA5 async copy: global memory -> LDS, 16 bytes per lane, ASYNCcnt-tracked.
// GV addressing: LDS byte address in the vdst VGPR, 64-bit global address in vaddr.
__device__ __forceinline__ void async_gather_b128(uint32_t lds_byte_addr,
                                                  const void* gaddr) {
  asm volatile("global_load_async_to_lds_b128 %0, %1, off"
               :
               : "v"(lds_byte_addr), "v"(gaddr)
               : "memory");
}

// Drain ASYNCcnt. Inline asm with a memory clobber so the optimizer cannot
// schedule the subsequent LDS reads above the wait (RAW on the staging tile).
__device__ __forceinline__ void wait_async0() {
  asm volatile("s_wait_asynccnt 0x0" ::: "memory");
}

__global__ __launch_bounds__(THREADS) void MeanAggregator_51642686767895_kernel(
    const float* __restrict__ features,
    const long long* __restrict__ neigh_idx,
    float* __restrict__ out,
    int n_nodes) {
  // Per-wave staging tile: NUM_SAMPLE rows x 128 floats = 5 KB; 8 waves -> 40 KB
  // (well under 320 KB/WGP; 8 blocks can still be resident per WGP).
  __shared__ float lds_buf[WAVES_PER_BLOCK * NUM_SAMPLE * D_FEAT];

  const int lane = threadIdx.x & 31;
  const int wave = threadIdx.x >> 5;
  const int node = blockIdx.x * WAVES_PER_BLOCK + wave;
  if (node >= n_nodes) return;  // wave-uniform; grid divides exactly anyway

  float* wbuf = &lds_buf[wave * (NUM_SAMPLE * D_FEAT)];
  // Low 32 bits of the generic address of a __shared__ object are the LDS
  // byte offset (aperture mapping discards the upper bits).
  const uint32_t wbuf_lds = (uint32_t)(uintptr_t)wbuf;

  // ---- Phase 1: load all indices first (no asm in between), so the
  // compiler can clause the 10 b64 loads and emit a single s_wait_loadcnt.
  const long long ibase = (long long)node * NUM_SAMPLE;
  long long idx[NUM_SAMPLE];
#pragma unroll
  for (int s = 0; s < NUM_SAMPLE; ++s) {
    idx[s] = __builtin_nontemporal_load(&neigh_idx[ibase + s]);
  }

  // ---- Phase 2: issue all 10 async gathers back-to-back; lane l copies
  // bytes [16l, 16l+16) of row idx[s] into the wave's staging tile.
  const float* fbase = features + (lane << 2);
  const uint32_t dbase = wbuf_lds + (uint32_t)(lane << 4);
#pragma unroll
  for (int s = 0; s < NUM_SAMPLE; ++s) {
    async_gather_b128(dbase + (uint32_t)(s * D_FEAT * 4),
                      (const void*)(fbase + idx[s] * D_FEAT));
  }

  // ---- Phase 3: drain, reduce from LDS, scale, NT-store.
  wait_async0();

  v4f acc = (v4f)(0.0f);
  const v4f* vbuf = (const v4f*)wbuf;
#pragma unroll
  for (int s = 0; s < NUM_SAMPLE; ++s) {
    acc += vbuf[s * (D_FEAT / 4) + lane];
  }
  acc *= (1.0f / (float)NUM_SAMPLE);

  // Non-temporal store: output is write-once; keep the feature table in L2.
  v4f* o = (v4f*)(out + (size_t)node * D_FEAT) + lane;
  __builtin_nontemporal_store(acc, o);
}

extern "C" void kernel_launch(void* const* d_in, const int* in_sizes, int n_in,
                              void* d_out, int out_size, void* d_ws, size_t ws_size,
                              hipStream_t stream) {
  const float* features = (const float*)d_in[0];
  const long long* neigh_idx = (const long long*)d_in[1];  // int64 per reference
  float* out = (float*)d_out;

  const int n_nodes = in_sizes[1] / NUM_SAMPLE;  // 100000
  const int blocks = (n_nodes + WAVES_PER_BLOCK - 1) / WAVES_PER_BLOCK;

  MeanAggregator_51642686767895_kernel<<<blocks, THREADS, 0, stream>>>(
      features, neigh_idx, out, n_nodes);
}